// BiLinearDecoder_8194797601146
// MI455X (gfx1250) — compile-verified
//
#include <hip/hip_runtime.h>

typedef float v2f __attribute__((ext_vector_type(2)));
typedef float v4f __attribute__((ext_vector_type(4)));
typedef float v8f __attribute__((ext_vector_type(8)));

namespace {
constexpr int kU = 1000, kV = 1000, kD = 128, kNW = 2, kNC = 2, kM = 16;
constexpr int kTU = 63, kTV = 63;        // ceil(1000/16)
constexpr int kPadRows = kTU * 16;       // 1008 padded rows for proj workspace
}

// ---------------------------------------------------------------------------
// Kernel A: proj[k] = lnc @ W[k]  -> padded (2 x 1008 x 128) fp32 workspace.
// One wave per (k, m-tile, n-tile); fp32 WMMA 16x16x4, 32 K-steps over d.
// lnc row reads clamped to row 999 for the partial last m-tile (pure VALU
// select, EXEC stays all-1s as WMMA requires).
// ---------------------------------------------------------------------------
__global__ __launch_bounds__(256) void proj_kernel(const float* __restrict__ feature,
                                                   const float* __restrict__ weight,
                                                   float* __restrict__ proj) {
  const int wave = blockIdx.x * 8 + (threadIdx.x >> 5);
  const int lane = threadIdx.x & 31;
  const int k  = wave / (kTU * 8);
  const int r0 = wave % (kTU * 8);
  const int mt = r0 >> 3;   // m-tile (0..62)
  const int nt = r0 & 7;    // n-tile (0..7), covers 128 output cols
  const int q  = lane >> 4; // half-wave -> K pair / M half
  const int n  = lane & 15;

  const float* lnc = feature;
  const float* W   = weight + (size_t)k * kD * kD;

  int arow = mt * 16 + n; arow = arow < kU ? arow : kU - 1;
  const float* aptr = lnc + (size_t)arow * kD + 2 * q;     // +4 per step
  const float* bptr = W + (size_t)(2 * q) * kD + nt * 16 + n; // +4*kD per step

  v8f acc = {};
#pragma unroll 4
  for (int d = 0; d < kD; d += 4) {
    v2f a = *(const v2f*)aptr;          // A[m][d+2q], A[m][d+2q+1]
    v2f b;                               // B[d+2q][n], B[d+2q+1][n]
    b.x = bptr[0];
    b.y = bptr[kD];
    acc = __builtin_amdgcn_wmma_f32_16x16x4_f32(false, a, false, b,
                                                (short)0, acc, false, false);
    aptr += 4;
    bptr += 4 * kD;
  }

  // C layout: VGPR r holds M = 8*q + r, N = lane&15
  float* out = proj + ((size_t)k * kPadRows + mt * 16) * kD + nt * 16 + n;
#pragma unroll
  for (int r = 0; r < 8; ++r)
    out[(size_t)(q * 8 + r) * kD] = acc[r];
}

// ---------------------------------------------------------------------------
// Kernel B: per 16x16 (u,v) tile: scores_k = proj_k[u-tile] @ dis[v-tile]^T
// (K=128, 32 fp32-WMMA steps per k), then fused classifier + ReLU epilogue.
// One wave per tile; B (dis rows) shared between the two k accumulators.
// ---------------------------------------------------------------------------
__global__ __launch_bounds__(256) void scores_kernel(const float* __restrict__ feature,
                                                     const float* __restrict__ proj,
                                                     const float* __restrict__ wc,
                                                     float* __restrict__ out) {
  const int t = blockIdx.x * 8 + (threadIdx.x >> 5);
  if (t >= kTU * kTV) return;           // wave-uniform exit, before any WMMA
  const int lane = threadIdx.x & 31;
  const int ut = t / kTV;
  const int vt = t % kTV;
  const int q = lane >> 4, n = lane & 15;

  const float* dis = feature + (size_t)kU * kD;
  int vrow = vt * 16 + n; vrow = vrow < kV ? vrow : kV - 1;

  const float* a0 = proj + ((size_t)0 * kPadRows + ut * 16 + n) * kD + 2 * q;
  const float* a1 = proj + ((size_t)1 * kPadRows + ut * 16 + n) * kD + 2 * q;
  const float* bp = dis + (size_t)vrow * kD + 2 * q;

  v8f c0 = {}, c1 = {};
#pragma unroll 4
  for (int e = 0; e < kD; e += 4) {
    v2f b  = *(const v2f*)bp;   // dis[v][e+2q], dis[v][e+2q+1]  (B col = dis row)
    v2f x0 = *(const v2f*)a0;
    v2f x1 = *(const v2f*)a1;
    c0 = __builtin_amdgcn_wmma_f32_16x16x4_f32(false, x0, false, b,
                                               (short)0, c0, false, false);
    c1 = __builtin_amdgcn_wmma_f32_16x16x4_f32(false, x1, false, b,
                                               (short)0, c1, false, false);
    a0 += 4; a1 += 4; bp += 4;
  }

  const float w00 = wc[0], w01 = wc[1], w10 = wc[2], w11 = wc[3];
#pragma unroll
  for (int r = 0; r < 8; ++r) {
    const int u = ut * 16 + q * 8 + r;
    const int v = vt * 16 + n;
    if (u < kU && v < kV) {
      const float s0 = c0[r], s1 = c1[r];
      float o0 = s0 * w00 + s1 * w10; o0 = o0 > 0.f ? o0 : 0.f;
      float o1 = s0 * w01 + s1 * w11; o1 = o1 > 0.f ? o1 : 0.f;
      v2f o; o.x = o0; o.y = o1;
      *(v2f*)(out + ((size_t)u * kV + v) * kNC) = o;   // 8-byte store
    }
  }
}

// ---------------------------------------------------------------------------
// Kernel C: the bandwidth-critical part (~1 GB of stores).
// grid = (125, 1000): blockIdx.y = u, 8 waves/block cover v in [0,1000).
// One wave per output row: each lane does one b128 load per source (L2-hot)
// and one non-temporal b128 store per destination (streaming, skip L2 rinse).
// ---------------------------------------------------------------------------
__global__ __launch_bounds__(256) void rep_kernel(const float* __restrict__ feature,
                                                  float* __restrict__ lnc_rep,
                                                  float* __restrict__ dis_rep) {
  const int u = blockIdx.y;
  const int v = blockIdx.x * 8 + (threadIdx.x >> 5);
  const int col = (threadIdx.x & 31) * 4;          // 32 lanes * 4 floats = 128
  const v4f lv = *(const v4f*)(feature + (size_t)u * kD + col);
  const v4f dv = *(const v4f*)(feature + ((size_t)kU + v) * kD + col);
  const size_t row = (size_t)u * kV + v;
  __builtin_nontemporal_store(lv, (v4f*)(lnc_rep + row * kD + col));
  __builtin_nontemporal_store(dv, (v4f*)(dis_rep + row * kD + col));
}

// ---------------------------------------------------------------------------
// Kernel D: passthrough copy of M (16 x 128 floats = 512 v4f).
// ---------------------------------------------------------------------------
__global__ __launch_bounds__(256) void mcopy_kernel(const float* __restrict__ feature,
                                                    float* __restrict__ mout) {
  const int i = blockIdx.x * 256 + threadIdx.x;
  const v4f* src = (const v4f*)(feature + (size_t)(kU + kV) * kD);
  if (i < (kM * kD) / 4)
    ((v4f*)mout)[i] = src[i];
}

extern "C" void kernel_launch(void* const* d_in, const int* in_sizes, int n_in,
                              void* d_out, int out_size, void* d_ws, size_t ws_size,
                              hipStream_t stream) {
  const float* feature = (const float*)d_in[0];   // (2016, 128) f32
  const float* weight  = (const float*)d_in[1];   // (2, 128, 128) f32
  const float* wc      = (const float*)d_in[2];   // (2, 2) f32

  float* out = (float*)d_out;
  float* outputs = out;                                      // (1e6, 2)
  float* lnc_rep = outputs + (size_t)kU * kV * kNC;          // (1e6, 128)
  float* dis_rep = lnc_rep + (size_t)kU * kV * kD;           // (1e6, 128)
  float* m_out   = dis_rep + (size_t)kU * kV * kD;           // (16, 128)

  float* proj = (float*)d_ws;   // 2 * 1008 * 128 floats ≈ 1 MB scratch

  // A: proj (2*63*8 = 1008 waves -> 126 blocks of 8 waves)
  proj_kernel<<<dim3(kNW * kTU), 256, 0, stream>>>(feature, weight, proj);
  // B: scores + classifier (63*63 = 3969 tiles, 8 waves/block)
  scores_kernel<<<dim3((kTU * kTV + 7) / 8), 256, 0, stream>>>(feature, proj, wc, outputs);
  // C: 1 GB of streaming replication stores (125*8 waves cover v exactly)
  rep_kernel<<<dim3(kV / 8, kU), 256, 0, stream>>>(feature, lnc_rep, dis_rep);
  // D: M copy
  mcopy_kernel<<<dim3(2), 256, 0, stream>>>(feature, m_out);
}